// MPNNNet_41180146434905
// MI455X (gfx1250) — compile-verified
//
#include <hip/hip_runtime.h>

#define NFEAT 128
#define NHID  128
#define NCLASS 40
#define NCP    48   // padded class dim for WMMA tiles (3 x 16)

typedef float v2f __attribute__((ext_vector_type(2)));
typedef float v8f __attribute__((ext_vector_type(8)));

// ---------------------------------------------------------------------------
// Layer-1 GEMM: h[i][j] = sum_k x[i][k] * W1[j][k] + b1[j]
// grid.x = nNodes/16 row-tiles; block = 256 threads = 8 waves; wave w owns
// N-tile w (cols 16w..16w+15). Uses V_WMMA_F32_16X16X4_F32, K-loop of 32.
// ---------------------------------------------------------------------------
__global__ __launch_bounds__(256) void linear1_wmma(
    const float* __restrict__ x, const float* __restrict__ W,
    const float* __restrict__ bias, float* __restrict__ h)
{
    const int wave = threadIdx.x >> 5;
    const int lane = threadIdx.x & 31;
    const int half = lane >> 4;      // 0: K pair {0,1}, 1: K pair {2,3}
    const int l16  = lane & 15;
    const int mBase = blockIdx.x << 4;
    const int nBase = wave << 4;

    // A fragment source: row (mBase+l16), K offset 2*half  (contiguous float2)
    const float* arow = x + (size_t)(mBase + l16) * NFEAT + 2 * half;
    // B fragment source: B[k][n] = W[n][k]; col (nBase+l16), K offset 2*half
    const float* brow = W + (size_t)(nBase + l16) * NFEAT + 2 * half;

    v8f acc = {0.f, 0.f, 0.f, 0.f, 0.f, 0.f, 0.f, 0.f};
#pragma unroll
    for (int k = 0; k < NFEAT; k += 4) {
        v2f a = *(const v2f*)(arow + k);
        v2f b = *(const v2f*)(brow + k);
        acc = __builtin_amdgcn_wmma_f32_16x16x4_f32(
            false, a, false, b, (short)0, acc, false, false);
    }

    const float bj = bias[nBase + l16];
    float* out = h + (size_t)(mBase + 8 * half) * NHID + nBase + l16;
#pragma unroll
    for (int r = 0; r < 8; ++r)
        out[(size_t)r * NHID] = acc[r] + bj;
}

// ---------------------------------------------------------------------------
// Layer-2 GEMM: outp[i][j] = sum_k h[i][k] * W2[j][k] + b2[j], j < 40.
// 3 waves/block (N-tiles 0..2 covering cols 0..47); cols >= 40 masked to 0.
// Output stored with padded stride NCP=48 (garbage cols never read).
// ---------------------------------------------------------------------------
__global__ __launch_bounds__(96) void linear2_wmma(
    const float* __restrict__ h, const float* __restrict__ W2,
    const float* __restrict__ b2, float* __restrict__ outp)
{
    const int wave = threadIdx.x >> 5;
    const int lane = threadIdx.x & 31;
    const int half = lane >> 4;
    const int l16  = lane & 15;
    const int mBase = blockIdx.x << 4;
    const int nBase = wave << 4;
    const int col   = nBase + l16;
    const bool valid = (col < NCLASS);

    const float* arow = h + (size_t)(mBase + l16) * NHID + 2 * half;
    const int ccl = valid ? col : (NCLASS - 1);
    const float* brow = W2 + (size_t)ccl * NHID + 2 * half;

    v8f acc = {0.f, 0.f, 0.f, 0.f, 0.f, 0.f, 0.f, 0.f};
#pragma unroll
    for (int k = 0; k < NHID; k += 4) {
        v2f a = *(const v2f*)(arow + k);
        v2f b = *(const v2f*)(brow + k);
        if (!valid) { b.x = 0.f; b.y = 0.f; }
        acc = __builtin_amdgcn_wmma_f32_16x16x4_f32(
            false, a, false, b, (short)0, acc, false, false);
    }

    const float bj = valid ? b2[col] : 0.f;
    float* out = outp + (size_t)(mBase + 8 * half) * NCP + col;
#pragma unroll
    for (int r = 0; r < 8; ++r)
        out[(size_t)r * NCP] = acc[r] + bj;
}

// ---------------------------------------------------------------------------
// Per-edge scatter for layer 1: 32 threads/edge, 4 feats/thread (float4 gather,
// 4x global_atomic_add_f32). Thread 0 of each edge also bumps the degree count.
// ---------------------------------------------------------------------------
__global__ void scatter1(const float* __restrict__ h,
                         const int* __restrict__ ei, int nEdges,
                         float* agg, float* cnt)
{
    long long i = (long long)blockIdx.x * blockDim.x + threadIdx.x;
    int e = (int)(i >> 5);
    int c = (int)(i & 31);
    if (e >= nEdges) return;
    int s = ei[e];
    int d = ei[nEdges + e];
    const float4 v = *(const float4*)(h + (size_t)s * NHID + 4 * c);
    float* a = agg + (size_t)d * NHID + 4 * c;
    atomicAdd(a + 0, v.x);
    atomicAdd(a + 1, v.y);
    atomicAdd(a + 2, v.z);
    atomicAdd(a + 3, v.w);
    if (c == 0) atomicAdd(cnt + d, 1.0f);
}

// Fused self-loop add + mean + ReLU (in place over agg1 -> becomes layer-2 input)
__global__ void mean_relu(const float* __restrict__ h1, float* agg,
                          const float* __restrict__ cnt, long long total)
{
    long long i = (long long)blockIdx.x * blockDim.x + threadIdx.x;
    if (i >= total) return;
    int node = (int)(i >> 7);   // /NHID
    float v = (agg[i] + h1[i]) / cnt[node];
    agg[i] = v > 0.f ? v : 0.f;
}

// Per-(edge,class) scatter for layer 2 into agg2 (= d_out used as accumulator)
__global__ void scatter2(const float* __restrict__ outp,
                         const int* __restrict__ ei, int nEdges,
                         float* agg2)
{
    long long i = (long long)blockIdx.x * blockDim.x + threadIdx.x;
    long long tot = (long long)nEdges * NCLASS;
    if (i >= tot) return;
    int e = (int)(i / NCLASS);
    int f = (int)(i - (long long)e * NCLASS);
    int s = ei[e];
    int d = ei[nEdges + e];
    atomicAdd(agg2 + (size_t)d * NCLASS + f, outp[(size_t)s * NCP + f]);
}

__global__ void set1f(float* p, int n)
{
    int i = blockIdx.x * blockDim.x + threadIdx.x;
    if (i < n) p[i] = 1.0f;
}

// Fused self-loop add + mean + log_softmax, in place over io (= d_out).
// Each thread reads its full 40-wide row into registers before writing back.
__global__ void finalize(float* io, const float* __restrict__ outp,
                         const float* __restrict__ cnt, int nNodes)
{
    int i = blockIdx.x * blockDim.x + threadIdx.x;
    if (i >= nNodes) return;
    const float inv = 1.0f / cnt[i];
    float v[NCLASS];
    float m = -3.402823466e38f;
#pragma unroll
    for (int f = 0; f < NCLASS; ++f) {
        float t = (io[(size_t)i * NCLASS + f] + outp[(size_t)i * NCP + f]) * inv;
        v[f] = t;
        m = fmaxf(m, t);
    }
    float s = 0.f;
#pragma unroll
    for (int f = 0; f < NCLASS; ++f) s += expf(v[f] - m);
    const float ls = logf(s);
#pragma unroll
    for (int f = 0; f < NCLASS; ++f)
        io[(size_t)i * NCLASS + f] = v[f] - m - ls;
}

extern "C" void kernel_launch(void* const* d_in, const int* in_sizes, int n_in,
                              void* d_out, int out_size, void* d_ws, size_t ws_size,
                              hipStream_t stream)
{
    const float* x  = (const float*)d_in[0];
    const int*   ei = (const int*)d_in[1];
    const float* W1 = (const float*)d_in[2];
    const float* b1 = (const float*)d_in[3];
    const float* W2 = (const float*)d_in[4];
    const float* b2 = (const float*)d_in[5];
    float* out = (float*)d_out;

    const int nNodes = in_sizes[0] / NFEAT;   // 100000
    const int nEdges = in_sizes[1] / 2;       // 1600000

    char* ws = (char*)d_ws;
    float* h1   = (float*)ws;                                        // nNodes*128 f32
    float* agg1 = (float*)(ws + (size_t)nNodes * NFEAT * 4);         // nNodes*128 f32
    float* cnt  = (float*)(ws + (size_t)nNodes * NFEAT * 8);         // nNodes f32
    float* outp = h1;   // reuse h1 region for padded layer-2 output (nNodes*48 f32)

    // Accumulator init (buffers are poisoned; must be re-zeroed every call)
    hipMemsetAsync(agg1, 0, (size_t)nNodes * NHID * sizeof(float), stream);
    hipMemsetAsync(out,  0, (size_t)nNodes * NCLASS * sizeof(float), stream);
    set1f<<<(nNodes + 255) / 256, 256, 0, stream>>>(cnt, nNodes);    // self-loop count

    // Layer 1: linear (WMMA f32) -> edge scatter-add -> mean+ReLU
    linear1_wmma<<<nNodes / 16, 256, 0, stream>>>(x, W1, b1, h1);
    {
        long long t = (long long)nEdges * 32;
        scatter1<<<(unsigned)((t + 255) / 256), 256, 0, stream>>>(h1, ei, nEdges, agg1, cnt);
    }
    {
        long long t = (long long)nNodes * NHID;
        mean_relu<<<(unsigned)((t + 255) / 256), 256, 0, stream>>>(h1, agg1, cnt, t);
    }

    // Layer 2: linear (WMMA f32, padded N) -> edge scatter-add -> mean+log_softmax
    linear2_wmma<<<nNodes / 16, 96, 0, stream>>>(agg1, W2, b2, outp);
    {
        long long t = (long long)nEdges * NCLASS;
        scatter2<<<(unsigned)((t + 255) / 256), 256, 0, stream>>>(outp, ei, nEdges, out);
    }
    finalize<<<(nNodes + 255) / 256, 256, 0, stream>>>(out, outp, cnt, nNodes);
}